// RoiPooling_79319456022579
// MI455X (gfx1250) — compile-verified
//
#include <hip/hip_runtime.h>
#include <cstdint>

// ---------------------------------------------------------------------------
// ROI max-pool for MI455X (gfx1250).
// 1 workgroup (8 wave32s) per ROI. The ROI's (<=28x28) feature window for a
// chunk of 8 channels is staged into LDS with CDNA5 async global->LDS loads
// (GLOBAL_LOAD_ASYNC_TO_LDS_B32, tracked by ASYNCcnt), double-buffered so the
// DMA of chunk i+1 overlaps the 7x7 masked-max compute of chunk i.
// Padding loads are branch-free (v_cndmask-selected sink) so every wave issues
// exactly LOADS_PER_THREAD async ops per chunk, making "s_wait_asynccnt 25"
// an exact chunk-boundary fence.
// ---------------------------------------------------------------------------

#define NTHREADS 256
#define MAXS 28                        // static max ROI extent (cells)
#define CCH 8                          // channels per LDS chunk
#define CHUNK_ELTS (CCH * MAXS * MAXS) // 6272 floats = 25,088 B
#define LOADS_PER_THREAD 25            // ceil(6272/256)

__device__ __forceinline__ void async_load_f32(uint32_t lds_off, uint64_t gaddr) {
    // vdst = LDS byte-address VGPR, vaddr = 64-bit global address (GV mode)
    asm volatile("global_load_async_to_lds_b32 %0, %1, off"
                 :: "v"(lds_off), "v"(gaddr)
                 : "memory");
}
__device__ __forceinline__ void wait_async_25() {
    asm volatile("s_wait_asynccnt 25" ::: "memory");
}
__device__ __forceinline__ void wait_async_0() {
    asm volatile("s_wait_asynccnt 0" ::: "memory");
}

__global__ __launch_bounds__(NTHREADS)
void roi_pool_async_kernel(const float* __restrict__ x,
                           const float* __restrict__ boxes,
                           float* __restrict__ out)
{
    constexpr int C = 256, H = 100, W = 152;
    constexpr float SCALE = 0.125f;

    __shared__ float sbuf[2][CHUNK_ELTS];
    __shared__ float sdummy[32];   // per-lane sink for the padding async loads

    const int k   = blockIdx.x;
    const int tid = threadIdx.x;

    // ---- per-ROI geometry (torchvision roi_pool quantization) ----
    const float* box = boxes + (size_t)k * 5;
    const int b  = (int)box[0];
    const int x1 = __float2int_rn(box[1] * SCALE);   // round-half-even == jnp.round
    const int y1 = __float2int_rn(box[2] * SCALE);
    const int x2 = __float2int_rn(box[3] * SCALE);
    const int y2 = __float2int_rn(box[4] * SCALE);
    const int roi_w = max(x2 - x1 + 1, 1);
    const int roi_h = max(y2 - y1 + 1, 1);
    const float bin_h = (float)roi_h / 7.0f;
    const float bin_w = (float)roi_w / 7.0f;
    const int y0 = min(max(y1, 0), H);
    const int x0 = min(max(x1, 0), W);

    const uint32_t sink_lds =
        (uint32_t)(uintptr_t)&sdummy[tid & 31];          // distinct per lane
    const uint64_t sink_g = (uint64_t)(uintptr_t)x;      // always-valid address

    // ---- branch-free async staging of one 8-channel window chunk ----
    auto issue_chunk = [&](int chunk, int sel) {
        const int cbase = chunk * CCH;
        for (int it = 0; it < LOADS_PER_THREAD; ++it) {
            const int j     = tid + it * NTHREADS;
            const bool live = (j < CHUNK_ELTS);
            const int  jj   = live ? j : 0;              // safe index for math
            const int cc  = jj / (MAXS * MAXS);
            const int e   = jj - cc * (MAXS * MAXS);
            const int r   = e / MAXS;
            const int col = e - r * MAXS;
            const int gr  = min(y0 + r,   H - 1);        // clamp -> in-bounds;
            const int gc  = min(x0 + col, W - 1);        // clamped slots unused
            const size_t eoff =
                (((size_t)(b * C + cbase + cc) * H + gr) * W + gc);
            const uint64_t gaddr =
                live ? (uint64_t)(uintptr_t)(x + eoff) : sink_g;
            const uint32_t lds_off =
                live ? (uint32_t)(uintptr_t)&sbuf[sel][jj] : sink_lds;
            async_load_f32(lds_off, gaddr);              // full-EXEC, no branch
        }
    };

    constexpr int NCHUNKS = C / CCH;   // 32
    issue_chunk(0, 0);

    for (int ch = 0; ch < NCHUNKS; ++ch) {
        const int sel = ch & 1;
        if (ch + 1 < NCHUNKS) {
            issue_chunk(ch + 1, sel ^ 1);  // overlap next chunk's DMA
            wait_async_25();               // <=25 outstanding => chunk ch landed
        } else {
            wait_async_0();
        }
        __syncthreads();                   // chunk ch visible to all 8 waves

        // ---- masked max over each of the 8x49 output bins ----
        for (int o = tid; o < CCH * 49; o += NTHREADS) {
            const int cc  = o / 49;
            const int bin = o - cc * 49;
            const int ph  = bin / 7;
            const int pw  = bin - ph * 7;

            const int hs = min(max((int)floorf((float)ph * bin_h) + y1, 0), H);
            const int he = min(max((int)ceilf((float)(ph + 1) * bin_h) + y1, 0), H);
            const int ws = min(max((int)floorf((float)pw * bin_w) + x1, 0), W);
            const int we = min(max((int)ceilf((float)(pw + 1) * bin_w) + x1, 0), W);

            float m = -INFINITY;
            const float* base = &sbuf[sel][cc * (MAXS * MAXS)];
            for (int r = hs; r < he; ++r) {
                const float* row = base + (r - y0) * MAXS - x0;
                for (int c2 = ws; c2 < we; ++c2)
                    m = fmaxf(m, row[c2]);
            }
            if (he <= hs || we <= ws) m = 0.0f;   // empty bin -> 0 (reference)

            out[((size_t)k * C + (ch * CCH + cc)) * 49 + bin] = m;
        }
        __syncthreads();  // all reads of sbuf[sel] done before it is reused
    }
}

extern "C" void kernel_launch(void* const* d_in, const int* in_sizes, int n_in,
                              void* d_out, int out_size, void* d_ws, size_t ws_size,
                              hipStream_t stream) {
    (void)n_in; (void)out_size; (void)d_ws; (void)ws_size;
    const float* x     = (const float*)d_in[0];   // (4, 256, 100, 152) f32
    const float* boxes = (const float*)d_in[1];   // (K, 5) f32
    float* out         = (float*)d_out;           // (K, 256, 7, 7) f32
    const int K = in_sizes[1] / 5;                // 2048
    roi_pool_async_kernel<<<K, NTHREADS, 0, stream>>>(x, boxes, out);
}